// Embedding_9646496547384
// MI455X (gfx1250) — compile-verified
//
#include <hip/hip_runtime.h>
#include <math.h>

typedef __attribute__((ext_vector_type(2))) float v2f;
typedef __attribute__((ext_vector_type(4))) float v4f;
typedef __attribute__((ext_vector_type(8))) float v8f;

#define NB 1024        // N
#define GG 64          // G
#define LL 16          // L
#define NDIM 256       // NODE_DIM
#define EDIM 16        // EDGE_DIM

static __device__ __forceinline__ float softplusf(float x) {
    // stable: max(x,0) + log1p(exp(-|x|))  (setup-only; full-precision libm ok)
    return fmaxf(x, 0.0f) + log1pf(expf(-fabsf(x)));
}
// fast sigmoid: v_exp_f32 + v_rcp_f32 (hot path, ~4 VALU ops)
static __device__ __forceinline__ float sigmoid_fast(float x) {
    return __builtin_amdgcn_rcpf(1.0f + __expf(-x));
}
static __device__ __forceinline__ float sigmoidf_(float x) {
    return 1.0f / (1.0f + expf(-x));
}
static __device__ __forceinline__ float gelu_exact(float x) {
    return 0.5f * x * (1.0f + erff(x * 0.7071067811865476f));
}
static __device__ __forceinline__ v8f zero_v8f() {
    v8f v;
#pragma unroll
    for (int i = 0; i < 8; ++i) v[i] = 0.0f;
    return v;
}

// ---------------------------------------------------------------------------
// Kernel 1: per-node FFN + layernorm -> nf ; also data_ligand -> workspace
// One 256-thread block per node (B*N = 2048 blocks).
// Note: for "same" nodes exp_t == type_exp*scaler, so ffn(exp_t) == the
// reference's where(same, nf2, nf1): one FFN suffices.
// ---------------------------------------------------------------------------
__global__ __launch_bounds__(256) void node_kernel(
    const float* __restrict__ x, const float* __restrict__ type_exp,
    const int* __restrict__ cell_types, const float* __restrict__ exp_scaler,
    const float* __restrict__ W1, const float* __restrict__ b1,
    const float* __restrict__ W2, const float* __restrict__ b2,
    const float* __restrict__ ct_emb, const float* __restrict__ ln_g,
    const float* __restrict__ ln_b, const int* __restrict__ lig_idx,
    float* __restrict__ nf_out, float* __restrict__ data_ligand)
{
    __shared__ float s_expt[GG];
    __shared__ float s_hid[4 * NDIM];
    __shared__ float s_red[256];

    const int node = blockIdx.x;          // b*N + n
    const int b    = node >> 10;
    const int tid  = threadIdx.x;

    const int ct  = cell_types[node];
    const int ct0 = cell_types[b << 10];
    const bool same = (ct == ct0);

    if (tid < GG) {
        float s  = exp_scaler[tid]; s = s * s;
        float te = type_exp[(size_t)node * GG + tid];
        float xx = x[(size_t)node * GG + tid];
        s_expt[tid] = (same ? te : (te + xx)) * s;
    }
    __syncthreads();

    // data_ligand[b,n,l] = sqrt(mean(group0) * mean(group1))
    if (tid < LL) {
        const int* li = lig_idx + tid * 4;
        float g0 = 0.5f * (s_expt[li[0]] + s_expt[li[1]]);
        float g1 = 0.5f * (s_expt[li[2]] + s_expt[li[3]]);
        data_ligand[(size_t)node * LL + tid] = sqrtf(g0 * g1);
    }

    // hidden: 1024 units, 4 per thread
#pragma unroll
    for (int jj = 0; jj < 4; ++jj) {
        const int j = tid + jj * 256;
        float acc = b1[j];
#pragma unroll
        for (int g = 0; g < GG; ++g)
            acc = fmaf(s_expt[g], W1[(size_t)g * 1024 + j], acc);
        s_hid[j] = gelu_exact(acc);
    }
    __syncthreads();

    // output channel c = tid
    const int c = tid;
    float o = b2[c];
#pragma unroll 8
    for (int h = 0; h < 1024; ++h)
        o = fmaf(s_hid[h], W2[(size_t)h * NDIM + c], o);
    o += ct_emb[(size_t)ct * NDIM + c];

    // layernorm over 256 channels
    s_red[tid] = o;
    __syncthreads();
    for (int s = 128; s > 0; s >>= 1) {
        if (tid < s) s_red[tid] += s_red[tid + s];
        __syncthreads();
    }
    const float mean = s_red[0] * (1.0f / 256.0f);
    __syncthreads();
    const float dv = o - mean;
    s_red[tid] = dv * dv;
    __syncthreads();
    for (int s = 128; s > 0; s >>= 1) {
        if (tid < s) s_red[tid] += s_red[tid + s];
        __syncthreads();
    }
    const float var = s_red[0] * (1.0f / 256.0f);
    const float rs  = rsqrtf(var + 1e-5f);
    nf_out[(size_t)node * NDIM + c] = dv * rs * ln_g[c] + ln_b[c];
}

// ---------------------------------------------------------------------------
// Kernel 2: edges (WMMA 16x16x4 f32 FFN 16->64->16) + emb1 + emb2.
// Each wave32 owns a tile of 16 consecutive (b,i,j) pairs; grid-stride loop.
// All cross-lane exchange is intra-wave LDS (in-order DS pipe, no barriers);
// all branches are wave-uniform so EXEC is all ones at every WMMA.
// ---------------------------------------------------------------------------
#define EWAVES 8
__global__ __launch_bounds__(256) void edge_kernel(
    const float* __restrict__ dist,
    const float* __restrict__ data_ligand,
    const float* __restrict__ ws,  const float* __restrict__ bs,
    const float* __restrict__ w1e, const float* __restrict__ b1e,
    const float* __restrict__ w2e, const float* __restrict__ b2e,
    const float* __restrict__ Wl1, const float* __restrict__ Wl2,
    float* __restrict__ edges, float* __restrict__ emb1, float* __restrict__ emb2,
    int numTiles)
{
    __shared__ float s_lig[EWAVES][16][16];   // [pair][l]
    __shared__ float s_hid[EWAVES][16][64];   // [pair][hidden]
    __shared__ float s_d[EWAVES][16];         // per-pair distance

    const int tid  = threadIdx.x;
    const int wave = tid >> 5;
    const int lane = tid & 31;
    const int row  = lane & 15;
    const int hi   = lane >> 4;               // lane half (K / M split)

    // per-lane ligand distance-embedding params (this lane covers l = hi*8..hi*8+7)
    float sw[8], sb[8];
#pragma unroll
    for (int k = 0; k < 8; ++k) {
        sw[k] = softplusf(ws[hi * 8 + k]);
        sb[k] = bs[hi * 8 + k];
    }
    // per-lane edge-embedding params, remapped for b128 stores:
    // lane owns (pair = lane>>1, e = (lane&1)*8 .. +7)
    const int epair = lane >> 1;
    const int ebase = (lane & 1) * 8;
    float swe1[8], be1v[8], swe2[8], be2v[8];
#pragma unroll
    for (int k = 0; k < 8; ++k) {
        swe1[k] = softplusf(w1e[ebase + k]);
        be1v[k] = b1e[ebase + k];
        swe2[k] = softplusf(w2e[ebase + k]);
        be2v[k] = b2e[ebase + k];
    }

    // B-fragments for Wl1 (16x64) and Wl2 (64x16), f32 WMMA 16x16x4 layout:
    //   lanes 0-15:  v0=B[4k+0][col], v1=B[4k+1][col]
    //   lanes 16-31: v0=B[4k+2][col], v1=B[4k+3][col]
    v2f w1f[4][4];   // [n-chunk][k-chunk]
#pragma unroll
    for (int nc = 0; nc < 4; ++nc)
#pragma unroll
        for (int kk = 0; kk < 4; ++kk) {
            const int kr  = kk * 4 + hi * 2;
            const int col = nc * 16 + row;
            w1f[nc][kk].x = Wl1[kr * 64 + col];
            w1f[nc][kk].y = Wl1[(kr + 1) * 64 + col];
        }
    v2f w2f[16];
#pragma unroll
    for (int kk = 0; kk < 16; ++kk) {
        const int kr = kk * 4 + hi * 2;
        w2f[kk].x = Wl2[kr * 16 + row];
        w2f[kk].y = Wl2[(kr + 1) * 16 + row];
    }

    const int gw         = blockIdx.x * EWAVES + wave;
    const int totalWaves = gridDim.x * EWAVES;

    for (int tile = gw; tile < numTiles; tile += totalWaves) {
        const int q0 = tile << 4;             // flat pair index of first pair
        const int bi = q0 >> 10;              // b*N + i   (j0..j0+15 same row)
        const int j0 = q0 & 1023;
        const int b  = bi >> 10;

        // distance of this lane's pair (pair = row); lanes 16-31 duplicate
        const float d = dist[(size_t)bi * NB + j0 + row];
        if (hi == 0) s_d[wave][row] = d;

        // lig_in tile: lane covers (pair=row, l=hi*8..hi*8+7)
        const float* dlp = data_ligand + ((size_t)(b * NB + j0 + row)) * LL + hi * 8;
#pragma unroll
        for (int k = 0; k < 8; ++k) {
            const float z = fmaf(d, sw[k], sb[k]);
            s_lig[wave][row][hi * 8 + k] = sigmoid_fast(z) * dlp[k];
        }

        // GEMM1: (16 pairs x 16 L) @ Wl1 (16 x 64) -> 16x64, K in 4 steps of 4
        v8f acc[4];
#pragma unroll
        for (int nc = 0; nc < 4; ++nc) acc[nc] = zero_v8f();
#pragma unroll
        for (int kk = 0; kk < 4; ++kk) {
            const v2f a = *(const v2f*)&s_lig[wave][row][kk * 4 + hi * 2];
#pragma unroll
            for (int nc = 0; nc < 4; ++nc)
                acc[nc] = __builtin_amdgcn_wmma_f32_16x16x4_f32(
                    false, a, false, w1f[nc][kk], (short)0, acc[nc], false, false);
        }

        // exact GELU, scatter C/D layout (M=r+8*hi, col=nc*16+row) into LDS as A source
#pragma unroll
        for (int nc = 0; nc < 4; ++nc)
#pragma unroll
            for (int r = 0; r < 8; ++r)
                s_hid[wave][r + hi * 8][nc * 16 + row] = gelu_exact(acc[nc][r]);

        // GEMM2: (16 pairs x 64) @ Wl2 (64 x 16) -> 16x16, K in 16 steps of 4
        v8f acc2 = zero_v8f();
#pragma unroll
        for (int kk = 0; kk < 16; ++kk) {
            const v2f a2 = *(const v2f*)&s_hid[wave][row][kk * 4 + hi * 2];
            acc2 = __builtin_amdgcn_wmma_f32_16x16x4_f32(
                false, a2, false, w2f[kk], (short)0, acc2, false, false);
        }

        // write edges (C/D layout: pair = r + 8*hi, e = row)
#pragma unroll
        for (int r = 0; r < 8; ++r) {
            const size_t q = (size_t)q0 + r + hi * 8;
            edges[q * EDIM + row] = acc2[r];
        }

        // emb1/emb2: lane owns (pair = epair, e = ebase..ebase+7) -> b128 stores
        {
            const float dp = s_d[wave][epair];
            v4f e1a, e1b, e2a, e2b;
#pragma unroll
            for (int k = 0; k < 4; ++k) {
                e1a[k] = sigmoid_fast(fmaf(dp, swe1[k],     be1v[k]));
                e1b[k] = sigmoid_fast(fmaf(dp, swe1[k + 4], be1v[k + 4]));
                e2a[k] = fmaf(dp, swe2[k],     be2v[k])     * 0.25f;   // 1/sqrt(16)
                e2b[k] = fmaf(dp, swe2[k + 4], be2v[k + 4]) * 0.25f;
            }
            const size_t q = (size_t)q0 + epair;
            *(v4f*)&emb1[q * EDIM + ebase]     = e1a;
            *(v4f*)&emb1[q * EDIM + ebase + 4] = e1b;
            *(v4f*)&emb2[q * EDIM + ebase]     = e2a;
            *(v4f*)&emb2[q * EDIM + ebase + 4] = e2b;
        }
    }
}

// ---------------------------------------------------------------------------
extern "C" void kernel_launch(void* const* d_in, const int* in_sizes, int n_in,
                              void* d_out, int out_size, void* d_ws, size_t ws_size,
                              hipStream_t stream) {
    const float* x          = (const float*)d_in[0];
    const float* type_exp   = (const float*)d_in[1];
    const float* dist       = (const float*)d_in[2];
    const int*   cell_types = (const int*)  d_in[3];
    const float* exp_scaler = (const float*)d_in[4];
    const float* W1         = (const float*)d_in[5];
    const float* b1         = (const float*)d_in[6];
    const float* W2         = (const float*)d_in[7];
    const float* b2         = (const float*)d_in[8];
    const float* ct_emb     = (const float*)d_in[9];
    const float* ln_g       = (const float*)d_in[10];
    const float* ln_b       = (const float*)d_in[11];
    const float* ws_        = (const float*)d_in[12];
    const float* bs_        = (const float*)d_in[13];
    const float* w1e        = (const float*)d_in[14];
    const float* b1e        = (const float*)d_in[15];
    const float* w2e        = (const float*)d_in[16];
    const float* b2e        = (const float*)d_in[17];
    const float* Wl1        = (const float*)d_in[18];
    const float* Wl2        = (const float*)d_in[19];
    const int*   lig_idx    = (const int*)  d_in[20];

    float* out = (float*)d_out;
    // outputs concatenated flat: nf, edges, emb1, emb2
    const size_t BN  = 2 * 1024;
    const size_t NN  = (size_t)2 * 1024 * 1024;
    float* nf_out   = out;
    float* edges_o  = out + BN * NDIM;                 // 524288
    float* emb1_o   = edges_o + NN * EDIM;             // +33554432
    float* emb2_o   = emb1_o  + NN * EDIM;

    float* data_ligand = (float*)d_ws;                 // B*N*L floats = 128 KB

    node_kernel<<<(int)BN, 256, 0, stream>>>(
        x, type_exp, cell_types, exp_scaler, W1, b1, W2, b2,
        ct_emb, ln_g, ln_b, lig_idx, nf_out, data_ligand);

    const int numTiles = (int)(NN / 16);               // 131072 16-pair tiles
    edge_kernel<<<2048, 256, 0, stream>>>(
        dist, data_ligand, ws_, bs_, w1e, b1e, w2e, b2e, Wl1, Wl2,
        edges_o, emb1_o, emb2_o, numTiles);
}